// RiemannianGeometryLayer_86242943304135
// MI455X (gfx1250) — compile-verified
//
#include <hip/hip_runtime.h>
#include <hip/hip_bf16.h>
#include <math.h>

// ---------------------------------------------------------------------------
// RiemannianGeometryLayer for MI455X (gfx1250, wave32, WMMA).
//
// Phase 1 (bandwidth-bound, dominates): per-batch covariance via
//   V_WMMA_F32_16X16X4_F32 Gram tiles. x is 360 MB fp32 -> ~15-30 us at
//   23.3 TB/s; FLOPs (2.1 G) are free under f32 WMMA. fp32 kept end-to-end
//   because Cholesky consumes the covariance.
// Phase 2: global trace-mean reduction (reference's Schoenberg reg couples
//   all batches, so it must be a separate kernel).
// Phase 3: 22x22 Cholesky (column-parallel in LDS) + MLP. Reference flattens
//   the *lower*-triangular log_cov with triu indices, so only the 22 diagonal
//   entries are nonzero -> the 253-dim GEMM collapses to a 22-row gather of W1.
// ---------------------------------------------------------------------------

typedef __attribute__((ext_vector_type(2))) float v2f;
typedef __attribute__((ext_vector_type(8))) float v8f;

constexpr int NC = 22;     // channels
constexpr int NT = 1000;   // time steps
constexpr int NB = 4096;   // batch
constexpr float K_REG = 1e-5f;
constexpr float K_SCH = 1e-3f;

// ---------------------------------------------------------------------------
// Phase 1: covariance. One block (4 waves) per batch element.
// Each wave owns every 4th K-step (step 4 floats per WMMA, stride 16).
// A(16x4) and B(4x16) f32 WMMA layouts coincide for Gram inputs, so a single
// float2 per lane feeds both operands. Row padding (22->32) is handled with a
// clamped pointer + 0/1 mask so the hot loop has no EXEC-mask manipulation.
// ---------------------------------------------------------------------------
__global__ __launch_bounds__(128) void cov_kernel(const float* __restrict__ x,
                                                  float* __restrict__ cov) {
    __shared__ float s_mean[NC];
    __shared__ float s_red[4 * 3 * 8 * 32];   // 4 waves x 3 tiles x 8 vgpr x 32 lanes

    const int b    = blockIdx.x;
    const int tid  = threadIdx.x;
    const int lane = tid & 31;
    const int w    = tid >> 5;
    const float* __restrict__ xb = x + (size_t)b * (NC * NT);

    // --- per-row means (coalesced: 32 lanes stride over T) ---
    for (int r = w; r < NC; r += 4) {
        float s = 0.f;
        for (int t = lane; t < NT; t += 32) s += xb[r * NT + t];
        for (int off = 16; off > 0; off >>= 1) s += __shfl_down(s, off, 32);
        if (lane == 0) s_mean[r] = s * (1.0f / NT);
    }
    __syncthreads();

    // --- Gram accumulation with V_WMMA_F32_16X16X4_F32 ---
    const int m   = lane & 15;             // row within 16-block (A) / col (B)
    const int kh  = lane >> 4;             // K half-select per ISA layout
    const int m2  = m + 16;                // second row block (rows 16..21 valid)
    const int m2c = (m2 < NC) ? m2 : (NC - 1);   // clamped row for safe loads
    const float msk  = (m2 < NC) ? 1.0f : 0.0f;  // zero out padded rows
    const float mu_a = s_mean[m];
    const float mu_c = s_mean[m2c];
    const float* __restrict__ pa = xb + m * NT;
    const float* __restrict__ pc = xb + m2c * NT;

    v8f t00 = {}; v8f t01 = {}; v8f t11 = {};
    #pragma unroll 2
    for (int k0 = w * 4; k0 < NT; k0 += 16) {
        const int k = k0 + 2 * kh;                 // even -> 8B aligned
        const float2 av = *(const float2*)(pa + k);
        const float2 cv = *(const float2*)(pc + k);
        v2f a, c;
        a.x = av.x - mu_a;
        a.y = av.y - mu_a;
        c.x = (cv.x - mu_c) * msk;                 // padded lanes contribute 0
        c.y = (cv.y - mu_c) * msk;
        // D = A x B + C ; Gram: same regs serve as A (16x4) and B (4x16)
        t00 = __builtin_amdgcn_wmma_f32_16x16x4_f32(false, a, false, a, (short)0, t00, false, false);
        t01 = __builtin_amdgcn_wmma_f32_16x16x4_f32(false, a, false, c, (short)0, t01, false, false);
        t11 = __builtin_amdgcn_wmma_f32_16x16x4_f32(false, c, false, c, (short)0, t11, false, false);
    }
    __syncthreads();

    // --- cross-wave tile reduction through LDS ---
    float* base = s_red + w * 768;
    #pragma unroll
    for (int v = 0; v < 8; ++v) {
        base[(0 * 8 + v) * 32 + lane] = t00[v];
        base[(1 * 8 + v) * 32 + lane] = t01[v];
        base[(2 * 8 + v) * 32 + lane] = t11[v];
    }
    __syncthreads();

    float* __restrict__ covb = cov + (size_t)b * (NC * NC);
    for (int p = tid; p < 768; p += 128) {
        const float s = s_red[p] + s_red[768 + p] + s_red[1536 + p] + s_red[2304 + p];
        const int t = p >> 8;            // tile: 0->(0,0) 1->(0,1) 2->(1,1)
        const int v = (p >> 5) & 7;      // C/D vgpr index
        const int l = p & 31;            // lane
        const int i = ((t == 2) ? 16 : 0) + v + ((l >> 4) << 3);  // M = v (+8 hi half)
        const int j = ((t == 0) ? 0 : 16) + (l & 15);             // N = lane%16
        if (i < NC && j < NC) {
            float val = s * (1.0f / (NT - 1));
            if (i == j) val += K_REG;                 // + REG * I (pre-trace)
            covb[i * NC + j] = val;
            if (t == 1) covb[j * NC + i] = val;       // mirror off-diag tile
        }
    }
}

// ---------------------------------------------------------------------------
// Phase 2: mean of traces across the whole batch -> scalar * SCHOENBERG_REG.
// ---------------------------------------------------------------------------
__global__ __launch_bounds__(256) void trace_kernel(const float* __restrict__ cov,
                                                    float* __restrict__ sreg) {
    __shared__ float part[256];
    float s = 0.f;
    for (int q = threadIdx.x; q < NB * NC; q += 256) {
        const int b = q / NC;
        const int i = q - b * NC;
        s += cov[(size_t)b * (NC * NC) + i * (NC + 1)];
    }
    part[threadIdx.x] = s;
    __syncthreads();
    for (int off = 128; off > 0; off >>= 1) {
        if (threadIdx.x < off) part[threadIdx.x] += part[threadIdx.x + off];
        __syncthreads();
    }
    if (threadIdx.x == 0) sreg[0] = (part[0] / (float)NB) * K_SCH;
}

// ---------------------------------------------------------------------------
// Phase 3: per-batch Cholesky (column-parallel, LDS, stride-23 padded) then
// the collapsed MLP: only diag(L)*log(diag(L)) survives the triu flatten.
// ---------------------------------------------------------------------------
__global__ __launch_bounds__(64) void chol_mlp_kernel(const float* __restrict__ cov,
                                                      const float* __restrict__ sreg,
                                                      const float* __restrict__ W1,
                                                      const float* __restrict__ b1,
                                                      const float* __restrict__ W2,
                                                      const float* __restrict__ b2,
                                                      float* __restrict__ out) {
    __shared__ float A[NC * 23];
    __shared__ float dlog[NC];
    __shared__ float h[256];

    const int b   = blockIdx.x;
    const int tid = threadIdx.x;
    const float* __restrict__ cb = cov + (size_t)b * (NC * NC);

    for (int q = tid; q < NC * NC; q += 64) {
        const int i = q / NC;
        const int j = q - i * NC;
        A[i * 23 + j] = cb[q];
    }
    __syncthreads();
    if (tid < NC) A[tid * 23 + tid] += sreg[0];     // Schoenberg regularizer
    __syncthreads();

    // Cholesky (lower), column-parallel: thread i owns row i.
    for (int j = 0; j < NC; ++j) {
        if (tid == j) A[j * 23 + j] = sqrtf(A[j * 23 + j]);
        __syncthreads();
        const float ljj = A[j * 23 + j];
        if (tid > j && tid < NC) A[tid * 23 + j] /= ljj;
        __syncthreads();
        if (tid > j && tid < NC) {
            const float lij = A[tid * 23 + j];
            for (int k = j + 1; k <= tid; ++k)
                A[tid * 23 + k] -= lij * A[k * 23 + j];
        }
        __syncthreads();
    }

    // Only diagonal of L*log(diag L) survives triu-flatten.
    if (tid < NC) {
        const float d = A[tid * 23 + tid];
        dlog[tid] = d * logf(fmaxf(d, 1e-10f));
    }
    __syncthreads();

    // h = elu(flattened @ W1 + b1): gather of 22 W1 rows at triu-diag indices.
    for (int n = tid; n < 256; n += 64) {
        float acc = b1[n];
        int ridx = 0;                               // index of (i,i) in triu order
        #pragma unroll
        for (int i = 0; i < NC; ++i) {
            acc = fmaf(dlog[i], W1[ridx * 256 + n], acc);
            ridx += NC - i;
        }
        h[n] = (acc > 0.f) ? acc : expm1f(acc);
    }
    __syncthreads();

    // out = h @ W2 + b2  (48 outputs)
    if (tid < 48) {
        float acc = b2[tid];
        for (int n = 0; n < 256; ++n)
            acc = fmaf(h[n], W2[n * 48 + tid], acc);
        out[(size_t)b * 48 + tid] = acc;
    }
}

// ---------------------------------------------------------------------------
extern "C" void kernel_launch(void* const* d_in, const int* in_sizes, int n_in,
                              void* d_out, int out_size, void* d_ws, size_t ws_size,
                              hipStream_t stream) {
    const float* x  = (const float*)d_in[0];
    const float* W1 = (const float*)d_in[1];
    const float* b1 = (const float*)d_in[2];
    const float* W2 = (const float*)d_in[3];
    const float* b2 = (const float*)d_in[4];
    float* out = (float*)d_out;

    float* cov  = (float*)d_ws;                         // NB*NC*NC floats (~7.9 MB)
    float* sreg = cov + (size_t)NB * NC * NC;           // 1 float

    cov_kernel<<<NB, 128, 0, stream>>>(x, cov);
    trace_kernel<<<1, 256, 0, stream>>>(cov, sreg);
    chol_mlp_kernel<<<NB, 64, 0, stream>>>(cov, sreg, W1, b1, W2, b2, out);
}